// LucasKanade_11897059409947
// MI455X (gfx1250) — compile-verified
//
#include <hip/hip_runtime.h>
#include <hip/hip_bf16.h>
#include <stdint.h>

#define IMG_W   1920
#define IMG_H   1080
#define IMG_C   3
#define PATCH   15
#define HALFP   7
#define NPIX    225     // 15*15
#define NSAMP   675     // 3*225
#define NPAD    676     // padded to multiple of 4 for WMMA K-chunks
#define TILE    16
#define N_STEPS 64
#define TOLF    1e-4f
#define EPSF    2.2204460492503131e-16f

typedef uint32_t u32x4 __attribute__((ext_vector_type(4)));
typedef uint32_t u32x8 __attribute__((ext_vector_type(8)));
typedef float    v2f   __attribute__((ext_vector_type(2)));
typedef float    v8f   __attribute__((ext_vector_type(8)));

// ---------------------------------------------------------------------------
// TDM: load a 16x16x3 float tile (x fastest, then y, then channel plane) from
// a CHW image into LDS using the Tensor Data Mover.
// ---------------------------------------------------------------------------
static __device__ __forceinline__ void tdm_load_tile(const float* img_base,
                                                     int bx, int by,
                                                     uint32_t lds_off) {
  uint64_t ga = (uint64_t)(uintptr_t)img_base +
                4ull * ((uint64_t)(uint32_t)by * (uint32_t)IMG_W + (uint32_t)bx);
  uint32_t ga_lo = (uint32_t)__builtin_amdgcn_readfirstlane((int)(ga & 0xffffffffull));
  uint32_t ga_hi = (uint32_t)__builtin_amdgcn_readfirstlane((int)(ga >> 32));
  uint32_t ldsa  = (uint32_t)__builtin_amdgcn_readfirstlane((int)lds_off);

  const uint32_t plane = (uint32_t)(IMG_W * IMG_H);     // 2073600
  u32x4 g0 = { 1u,                                      // count=1 (valid)
               ldsa,                                    // lds_addr
               ga_lo,                                   // global_addr[31:0]
               (ga_hi & 0x01ffffffu) | 0x80000000u };   // ga[56:32] | type=2
  u32x8 g1 = { 0x00020000u,                             // data_size=4B
               ((uint32_t)IMG_W & 0xffffu) << 16,       // tensor_dim0 lo
               ((uint32_t)IMG_H & 0xffffu) << 16,       // dim0 hi=0 | tensor_dim1 lo
               (uint32_t)TILE << 16,                    // dim1 hi=0 | tile_dim0=16
               ((uint32_t)IMG_C << 16) | (uint32_t)TILE,// tile_dim1=16 tile_dim2=3
               (uint32_t)IMG_W,                         // dim0_stride[31:0]
               (plane & 0xffffu) << 16,                 // d0s hi=0 | dim1_stride lo
               plane >> 16 };                           // dim1_stride[47:16]
  u32x4 g2 = { (uint32_t)IMG_C, 0u, 0u, 0u };           // tensor_dim2=3
  u32x4 g3 = { 0u, 0u, 0u, 0u };

  asm volatile("tensor_load_to_lds %0, %1, %2, %3"
               :: "s"(g0), "s"(g1), "s"(g2), "s"(g3)
               : "memory");
  __builtin_amdgcn_s_wait_tensorcnt(0);
}

static __device__ __forceinline__ float wave_sum(float v) {
  v += __shfl_xor(v, 16, 32);
  v += __shfl_xor(v, 8, 32);
  v += __shfl_xor(v, 4, 32);
  v += __shfl_xor(v, 2, 32);
  v += __shfl_xor(v, 1, 32);
  return v;
}

// Bilinear sample of patch element (c,i,j) from the LDS tile, with the exact
// per-tap validity masking of the reference (out-of-image taps contribute 0).
static __device__ __forceinline__ float sample_tile(const float* tile,
                                                    int c, int i, int j,
                                                    int ix, int iy,
                                                    int bx, int by,
                                                    float wx, float wy) {
  int x0 = ix - HALFP + j, x1 = x0 + 1;
  int y0 = iy - HALFP + i, y1 = y0 + 1;
  float vx0 = (x0 >= 0 && x0 < IMG_W) ? 1.f : 0.f;
  float vx1 = (x1 >= 0 && x1 < IMG_W) ? 1.f : 0.f;
  float vy0 = (y0 >= 0 && y0 < IMG_H) ? 1.f : 0.f;
  float vy1 = (y1 >= 0 && y1 < IMG_H) ? 1.f : 0.f;
  int tx0 = min(max(x0 - bx, 0), TILE - 1);
  int tx1 = min(max(x1 - bx, 0), TILE - 1);
  int ty0 = min(max(y0 - by, 0), TILE - 1);
  int ty1 = min(max(y1 - by, 0), TILE - 1);
  const float* tc = tile + c * (TILE * TILE);
  return tc[ty0 * TILE + tx0] * ((1.f - wy) * (1.f - wx) * vy0 * vx0)
       + tc[ty0 * TILE + tx1] * ((1.f - wy) * wx * vy0 * vx1)
       + tc[ty1 * TILE + tx0] * (wy * (1.f - wx) * vy1 * vx0)
       + tc[ty1 * TILE + tx1] * (wy * wx * vy1 * vx1);
}

__global__ __launch_bounds__(256)
void lk_track_kernel(const float* __restrict__ img0,
                     const float* __restrict__ img1,
                     const float* __restrict__ pts,
                     float* __restrict__ out, int P) {
  __shared__ float tile[IMG_C * TILE * TILE];   // 768 floats, TDM destination
  __shared__ float patch0[NSAMP];
  __shared__ float wgx[NPAD];                   // Gaussian-weighted grad x (A row 0)
  __shared__ float wgy[NPAD];                   // Gaussian-weighted grad y (A row 1)
  __shared__ float jx[NPAD];                    // unweighted grad x (B col 0)
  __shared__ float jy[NPAD];                    // unweighted grad y (B col 1)
  __shared__ float red[16];
  __shared__ float st[8];  // 0:cx 1:cy 2..5:invH 6:done

  const int p = blockIdx.x;
  if (p >= P) return;
  const int tid  = threadIdx.x;
  const int lane = tid & 31;
  const int wv   = tid >> 5;

  const float px = pts[2 * p + 0];
  const float py = pts[2 * p + 1];
  const float fx = floorf(px), fy = floorf(py);
  const float wx = px - fx,    wy = py - fy;
  const int ix = (int)fx, iy = (int)fy;
  const int bx = min(max(ix - HALFP, 0), IMG_W - TILE);
  const int by = min(max(iy - HALFP, 0), IMG_H - TILE);

  // Generic pointer to a __shared__ object: low 32 bits == LDS byte offset.
  const uint32_t tile_lds = (uint32_t)(uintptr_t)(void*)tile;

  // ---- Setup: frame0 tile -> patch0 -------------------------------------
  if (tid < 32) tdm_load_tile(img0, bx, by, tile_lds);
  __syncthreads();

  for (int n = tid; n < NSAMP; n += 256) {
    int c = n / NPIX; int r = n - c * NPIX;
    int i = r / PATCH; int j = r - i * PATCH;
    patch0[n] = sample_tile(tile, c, i, j, ix, iy, bx, by, wx, wy);
  }
  __syncthreads();

  // ---- Sobel (XLA correlation, zero pad) + Gaussian weighting -----------
  for (int n = tid; n < NSAMP; n += 256) {
    int c = n / NPIX; int r = n - c * NPIX;
    int i = r / PATCH; int j = r - i * PATCH;
    const float* pc = patch0 + c * NPIX;
    auto at = [&](int ii, int jj) -> float {
      return (ii < 0 || ii >= PATCH || jj < 0 || jj >= PATCH)
                 ? 0.f : pc[ii * PATCH + jj];
    };
    float gx = (-at(i - 1, j - 1) + at(i - 1, j + 1)
                - 2.f * at(i, j - 1) + 2.f * at(i, j + 1)
                - at(i + 1, j - 1) + at(i + 1, j + 1)) * 0.125f;
    float gy = (-at(i - 1, j - 1) - 2.f * at(i - 1, j) - at(i - 1, j + 1)
                + at(i + 1, j - 1) + 2.f * at(i + 1, j) + at(i + 1, j + 1)) * 0.125f;
    float di = (float)(i - HALFP), dj = (float)(j - HALFP);
    float g = (i == 0 || i == PATCH - 1 || j == 0 || j == PATCH - 1)
                  ? 0.f
                  : expf(-(di * di + dj * dj) * (1.f / 112.5f));
    jx[n]  = gx;      jy[n]  = gy;
    wgx[n] = g * gx;  wgy[n] = g * gy;
  }
  if (tid == 0) { wgx[NSAMP] = 0.f; wgy[NSAMP] = 0.f; jx[NSAMP] = 0.f; jy[NSAMP] = 0.f; }
  __syncthreads();

  // ---- Hessian H = wJ (2x675) * J^T (675x2) on the matrix unit ----------
  // V_WMMA_F32_16X16X4_F32, accumulated over 169 K=4 chunks.
  // A (16x4): lanes 0-15 M=0-15 hold K={0,1}; lanes 16-31 hold K={2,3}.
  //   row 0 = wgx, row 1 = wgy, rows 2-15 = 0.
  // B (4x16): lanes 0-15 N=0-15 hold K={0,1}; lanes 16-31 hold K={2,3}.
  //   col 0 = jx, col 1 = jy, cols 2-15 = 0.
  // D (16x16 f32): H00=D[0][0], H01=D[0][1], H11=D[1][1] -> VGPR0/1 lanes 0/1.
  if (tid < 32) {
    const int m  = lane & 15;            // M for A rows, N for B cols
    const int kb = (lane >> 4) << 1;     // 0 (lanes 0-15) or 2 (lanes 16-31)
    v8f acc = {};
    for (int nb = 0; nb < NSAMP; nb += 4) {
      int n0 = nb + kb;
      v2f a = { 0.f, 0.f };
      v2f b = { 0.f, 0.f };
      if (m == 0) {
        a.x = wgx[n0]; a.y = wgx[n0 + 1];
        b.x = jx[n0];  b.y = jx[n0 + 1];
      } else if (m == 1) {
        a.x = wgy[n0]; a.y = wgy[n0 + 1];
        b.x = jy[n0];  b.y = jy[n0 + 1];
      }
      acc = __builtin_amdgcn_wmma_f32_16x16x4_f32(
          false, a, false, b, (short)0, acc, false, false);
    }
    float H00 = __shfl(acc[0], 0, 32);
    float H01 = __shfl(acc[0], 1, 32);
    float H11 = __shfl(acc[1], 1, 32);
    if (lane == 0) {
      float a2 = H00 + EPSF, d2 = H11 + EPSF, b2 = H01;
      float det = a2 * d2 - b2 * b2 + EPSF;
      float inv = 1.f / det;
      st[2] = d2 * inv;  st[3] = -b2 * inv;
      st[4] = -b2 * inv; st[5] = a2 * inv;
      st[0] = px; st[1] = py; st[6] = 0.f;
    }
  }
  __syncthreads();

  // ---- 64 Gauss-Newton steps -------------------------------------------
  for (int it = 0; it < N_STEPS; ++it) {
    const float cx = st[0], cy = st[1];
    if (st[6] != 0.f) break;     // uniform across block (read after barrier)

    const float cfx = floorf(cx), cfy = floorf(cy);
    const float cwx = cx - cfx,  cwy = cy - cfy;
    const int cix = (int)cfx, ciy = (int)cfy;
    const int cbx = min(max(cix - HALFP, 0), IMG_W - TILE);
    const int cby = min(max(ciy - HALFP, 0), IMG_H - TILE);

    if (tid < 32) tdm_load_tile(img1, cbx, cby, tile_lds);
    __syncthreads();

    float sx = 0.f, sy = 0.f;
    for (int n = tid; n < NSAMP; n += 256) {
      int c = n / NPIX; int r = n - c * NPIX;
      int i = r / PATCH; int j = r - i * PATCH;
      float v = sample_tile(tile, c, i, j, cix, ciy, cbx, cby, cwx, cwy);
      float res = v - patch0[n];
      sx += wgx[n] * res;
      sy += wgy[n] * res;
    }
    sx = wave_sum(sx); sy = wave_sum(sy);
    if (lane == 0) { red[2 * wv] = sx; red[2 * wv + 1] = sy; }
    __syncthreads();
    if (tid == 0) {
      float SX = 0.f, SY = 0.f;
      for (int w = 0; w < 8; ++w) { SX += red[2 * w]; SY += red[2 * w + 1]; }
      float dx = st[2] * SX + st[3] * SY;
      float dy = st[4] * SX + st[5] * SY;
      st[0] = cx - dx;
      st[1] = cy - dy;
      if (fabsf(dx) < TOLF && fabsf(dy) < TOLF) st[6] = 1.f;
    }
    __syncthreads();
  }

  if (tid == 0) {
    out[2 * p + 0] = st[0];
    out[2 * p + 1] = st[1];
  }
}

extern "C" void kernel_launch(void* const* d_in, const int* in_sizes, int n_in,
                              void* d_out, int out_size, void* d_ws, size_t ws_size,
                              hipStream_t stream) {
  const float* img0 = (const float*)d_in[0];   // (1,3,1080,1920) f32
  const float* img1 = (const float*)d_in[1];   // (1,3,1080,1920) f32
  const float* pts  = (const float*)d_in[2];   // (1,4096,2) f32
  float* out = (float*)d_out;                  // (1,4096,2) f32
  const int P = in_sizes[2] / 2;
  lk_track_kernel<<<P, 256, 0, stream>>>(img0, img1, pts, out, P);
}